// Qeq_38225208934575
// MI455X (gfx1250) — compile-verified
//
#include <hip/hip_runtime.h>
#include <hip/hip_bf16.h>
#include <math.h>

#define QN      512          // atoms per molecule
#define QM      64           // molecules
#define QFEAT   128
#define NAUG    528          // 513 padded up to 33*16
#define NB      16
#define NBLK    (NAUG / NB)  // 33
#define LDA     NAUG

#define FACTOR   14.399645478425668f
#define SQRT_PI  1.7724538509055159f
#define SQRT2    1.4142135623730951f

typedef __attribute__((ext_vector_type(2))) float v2f;
typedef __attribute__((ext_vector_type(8))) float v8f;

// ---------------------------------------------------------------------------
// chi[i] = dot(node_features[i,:], w_chi)   (one wave32 per atom)
// ---------------------------------------------------------------------------
__global__ __launch_bounds__(256) void qeq_chi_kernel(
    const float* __restrict__ nf, const float* __restrict__ w,
    float* __restrict__ chi, int N)
{
    int wave = threadIdx.x >> 5;
    int lane = threadIdx.x & 31;
    int atom = blockIdx.x * 8 + wave;
    if (atom >= N) return;
    const float* row = nf + (size_t)atom * QFEAT;
    float s = 0.0f;
#pragma unroll
    for (int f = lane; f < QFEAT; f += 32) s += row[f] * w[f];
#pragma unroll
    for (int off = 16; off > 0; off >>= 1) s += __shfl_down(s, off, 32);
    if (lane == 0) chi[atom] = s;
}

// ---------------------------------------------------------------------------
// Build the padded augmented matrix A_aug (NAUG x NAUG) per molecule.
// rows/cols 0..511: Coulomb matrix; row/col 512: constraint; 513..527: identity pad
// ---------------------------------------------------------------------------
__global__ __launch_bounds__(256) void qeq_build_A_kernel(
    const float* __restrict__ pos, const float* __restrict__ hp,
    const float* __restrict__ st, const int* __restrict__ spec,
    float* __restrict__ Aall)
{
    int m = blockIdx.y;
    int idx = blockIdx.x * blockDim.x + threadIdx.x;
    if (idx >= NAUG * NAUG) return;
    int i = idx / NAUG;
    int j = idx - i * NAUG;
    float v;
    if (i < QN && j < QN) {
        int gi = m * QN + i;
        int gj = m * QN + j;
        if (i == j) {
            int s = spec[gi];
            float sg = st[s];
            float h = hp[s];
            v = h * h + FACTOR / (SQRT_PI * sg);
        } else {
            float dx = pos[3 * gi + 0] - pos[3 * gj + 0];
            float dy = pos[3 * gi + 1] - pos[3 * gj + 1];
            float dz = pos[3 * gi + 2] - pos[3 * gj + 2];
            float d = sqrtf(dx * dx + dy * dy + dz * dz);
            float si = st[spec[gi]];
            float sj = st[spec[gj]];
            float gma = sqrtf(si * si + sj * sj);
            v = FACTOR * erff(d / (SQRT2 * gma)) / d;
        }
    } else if (i == QN && j < QN) {
        v = 1.0f;
    } else if (j == QN && i < QN) {
        v = 1.0f;
    } else {
        v = (i == j && i > QN) ? 1.0f : 0.0f;  // (512,512)=0, pad diag=1
    }
    Aall[(size_t)m * NAUG * LDA + (size_t)i * LDA + j] = v;
}

// ---------------------------------------------------------------------------
// rhs = [-chi_m ; Q_m ; zeros pad]
// ---------------------------------------------------------------------------
__global__ __launch_bounds__(256) void qeq_build_rhs_kernel(
    const float* __restrict__ chi, const float* __restrict__ Qt,
    float* __restrict__ rhs)
{
    int m = blockIdx.y;
    int i = blockIdx.x * blockDim.x + threadIdx.x;
    if (i >= NAUG) return;
    float v = 0.0f;
    if (i < QN)       v = -chi[m * QN + i];
    else if (i == QN) v = Qt[m];
    rhs[m * NAUG + i] = v;
}

// ---------------------------------------------------------------------------
// Per-molecule blocked LU (no pivot, NB=16), trailing update via
// V_WMMA_F32_16X16X4_F32 with 2x2 register tile blocking per wave,
// then forward/backward substitution + energy.
// One workgroup (256 threads = 8 waves) per molecule.
// ---------------------------------------------------------------------------
__global__ __launch_bounds__(256) void qeq_lu_solve_kernel(
    float* __restrict__ Aall, const float* __restrict__ rhs,
    const float* __restrict__ chi, const float* __restrict__ Qt,
    float* __restrict__ out_q, float* __restrict__ out_e)
{
    __shared__ float Ds[NB][NB + 1];   // diagonal block (LU-factored in place)
    __shared__ float LpT[NB][520];     // NEGATED, transposed column panel
    __shared__ float xs[NAUG];
    __shared__ float red[256];

    int m = blockIdx.x;
    float* A = Aall + (size_t)m * NAUG * LDA;
    int tid  = threadIdx.x;
    int lane = tid & 31;
    int wave = tid >> 5;
    int n    = lane & 15;   // N/M index inside fragments
    int hi   = lane >> 4;   // lane half select

    for (int k = 0; k < NBLK; ++k) {
        int kb  = k * NB;
        int off = kb + NB;
        int rem = NAUG - off;   // always a multiple of 16

        // -- load diagonal block (256 threads = exactly 16x16)
        {
            int di = tid >> 4, dj = tid & 15;
            Ds[di][dj] = A[(size_t)(kb + di) * LDA + (kb + dj)];
        }
        __syncthreads();
        // -- factor 16x16 block serially (tiny)
        if (tid == 0) {
            for (int j = 0; j < NB; ++j) {
                float piv = Ds[j][j];
                for (int i2 = j + 1; i2 < NB; ++i2) {
                    float l = Ds[i2][j] / piv;
                    Ds[i2][j] = l;
                    for (int t = j + 1; t < NB; ++t) Ds[i2][t] -= l * Ds[j][t];
                }
            }
        }
        __syncthreads();
        // -- write back factored diagonal block
        {
            int di = tid >> 4, dj = tid & 15;
            A[(size_t)(kb + di) * LDA + (kb + dj)] = Ds[di][dj];
        }
        // -- column panel: L21 = A21 * inv(U11)  (one thread per row)
        for (int r = tid; r < rem; r += 256) {
            float* Arow = A + (size_t)(off + r) * LDA + kb;
            float a[NB];
#pragma unroll
            for (int j = 0; j < NB; ++j) a[j] = Arow[j];
#pragma unroll
            for (int j = 0; j < NB; ++j) {
                float s = a[j];
                for (int t = 0; t < j; ++t) s -= a[t] * Ds[t][j];
                a[j] = s / Ds[j][j];
            }
#pragma unroll
            for (int j = 0; j < NB; ++j) { Arow[j] = a[j]; LpT[j][r] = -a[j]; }
        }
        // -- row panel: U12 = inv(L11) * A12  (one thread per column)
        for (int c = tid; c < rem; c += 256) {
            int gc = off + c;
            float b[NB];
#pragma unroll
            for (int t = 0; t < NB; ++t) b[t] = A[(size_t)(kb + t) * LDA + gc];
#pragma unroll
            for (int t = 0; t < NB; ++t) {
                float s = b[t];
                for (int s2 = 0; s2 < t; ++s2) s -= Ds[t][s2] * b[s2];
                b[t] = s;
            }
#pragma unroll
            for (int t = 0; t < NB; ++t) A[(size_t)(kb + t) * LDA + gc] = b[t];
        }
        __syncthreads();
        // -- trailing update: A22 += (-L21) * U12, fp32 WMMA, 2x2 tiles/wave
        if (rem > 0) {
            int nT = rem >> 4;            // 16x16 tiles per dim
            int nS = (nT + 1) >> 1;       // 2x2 super-tiles per dim
            int totalS = nS * nS;
            for (int s = wave; s < totalS; s += 8) {   // wave-uniform
                int si = s / nS, sj = s - si * nS;
                int ti0 = si * 2, tj0 = sj * 2;
                bool di1 = (ti0 + 1) < nT;             // uniform per wave
                bool dj1 = (tj0 + 1) < nT;
                const float* Brow0 = A + (size_t)kb * LDA + (off + tj0 * NB);
                const float* Brow1 = Brow0 + NB;
                v2f a0[4], a1[4], b0[4], b1[4];
#pragma unroll
                for (int kk = 0; kk < 4; ++kk) {
                    int kr = 4 * kk + 2 * hi;
                    a0[kk][0] = LpT[kr + 0][ti0 * NB + n];
                    a0[kk][1] = LpT[kr + 1][ti0 * NB + n];
                    b0[kk][0] = Brow0[(size_t)(kr + 0) * LDA + n];
                    b0[kk][1] = Brow0[(size_t)(kr + 1) * LDA + n];
                }
                if (di1) {
#pragma unroll
                    for (int kk = 0; kk < 4; ++kk) {
                        int kr = 4 * kk + 2 * hi;
                        a1[kk][0] = LpT[kr + 0][(ti0 + 1) * NB + n];
                        a1[kk][1] = LpT[kr + 1][(ti0 + 1) * NB + n];
                    }
                }
                if (dj1) {
#pragma unroll
                    for (int kk = 0; kk < 4; ++kk) {
                        int kr = 4 * kk + 2 * hi;
                        b1[kk][0] = Brow1[(size_t)(kr + 0) * LDA + n];
                        b1[kk][1] = Brow1[(size_t)(kr + 1) * LDA + n];
                    }
                }
                float* C00 = A + (size_t)(off + ti0 * NB) * LDA + (off + tj0 * NB);
                float* C01 = C00 + NB;
                float* C10 = C00 + (size_t)NB * LDA;
                float* C11 = C10 + NB;
                v8f c00, c01, c10, c11;
#pragma unroll
                for (int r = 0; r < 8; ++r) c00[r] = C00[(size_t)(r + 8 * hi) * LDA + n];
                if (dj1) {
#pragma unroll
                    for (int r = 0; r < 8; ++r) c01[r] = C01[(size_t)(r + 8 * hi) * LDA + n];
                }
                if (di1) {
#pragma unroll
                    for (int r = 0; r < 8; ++r) c10[r] = C10[(size_t)(r + 8 * hi) * LDA + n];
                }
                if (di1 && dj1) {
#pragma unroll
                    for (int r = 0; r < 8; ++r) c11[r] = C11[(size_t)(r + 8 * hi) * LDA + n];
                }
#pragma unroll
                for (int kk = 0; kk < 4; ++kk)
                    c00 = __builtin_amdgcn_wmma_f32_16x16x4_f32(
                        false, a0[kk], false, b0[kk], (short)0, c00, false, false);
                if (dj1) {
#pragma unroll
                    for (int kk = 0; kk < 4; ++kk)
                        c01 = __builtin_amdgcn_wmma_f32_16x16x4_f32(
                            false, a0[kk], false, b1[kk], (short)0, c01, false, false);
                }
                if (di1) {
#pragma unroll
                    for (int kk = 0; kk < 4; ++kk)
                        c10 = __builtin_amdgcn_wmma_f32_16x16x4_f32(
                            false, a1[kk], false, b0[kk], (short)0, c10, false, false);
                }
                if (di1 && dj1) {
#pragma unroll
                    for (int kk = 0; kk < 4; ++kk)
                        c11 = __builtin_amdgcn_wmma_f32_16x16x4_f32(
                            false, a1[kk], false, b1[kk], (short)0, c11, false, false);
                }
#pragma unroll
                for (int r = 0; r < 8; ++r) C00[(size_t)(r + 8 * hi) * LDA + n] = c00[r];
                if (dj1) {
#pragma unroll
                    for (int r = 0; r < 8; ++r) C01[(size_t)(r + 8 * hi) * LDA + n] = c01[r];
                }
                if (di1) {
#pragma unroll
                    for (int r = 0; r < 8; ++r) C10[(size_t)(r + 8 * hi) * LDA + n] = c10[r];
                }
                if (di1 && dj1) {
#pragma unroll
                    for (int r = 0; r < 8; ++r) C11[(size_t)(r + 8 * hi) * LDA + n] = c11[r];
                }
            }
        }
        __syncthreads();
    }

    // ---- substitution: L y = b (unit lower), U x = y ----
    for (int i = tid; i < NAUG; i += 256) xs[i] = rhs[m * NAUG + i];
    __syncthreads();
    for (int i = 0; i < NAUG; ++i) {            // forward, column sweep
        float xi = xs[i];
        for (int j = i + 1 + tid; j < NAUG; j += 256)
            xs[j] -= A[(size_t)j * LDA + i] * xi;
        __syncthreads();
    }
    for (int i = NAUG - 1; i >= 0; --i) {       // backward
        if (tid == 0) xs[i] /= A[(size_t)i * LDA + i];
        __syncthreads();
        float xi = xs[i];
        for (int j = tid; j < i; j += 256)
            xs[j] -= A[(size_t)j * LDA + i] * xi;
        __syncthreads();
    }

    // ---- outputs: q, and e = 0.5*chi.q - 0.5*lambda*Q ----
    float lam = xs[QN];
    float s = 0.0f;
    for (int i = tid; i < QN; i += 256) {
        out_q[m * QN + i] = xs[i];
        s += chi[m * QN + i] * xs[i];
    }
    red[tid] = s;
    __syncthreads();
    for (int str = 128; str > 0; str >>= 1) {
        if (tid < str) red[tid] += red[tid + str];
        __syncthreads();
    }
    if (tid == 0) out_e[m] = 0.5f * red[0] - 0.5f * lam * Qt[m];
}

// ---------------------------------------------------------------------------
extern "C" void kernel_launch(void* const* d_in, const int* in_sizes, int n_in,
                              void* d_out, int out_size, void* d_ws, size_t ws_size,
                              hipStream_t stream) {
    (void)in_sizes; (void)n_in; (void)out_size; (void)ws_size;
    const float* pos  = (const float*)d_in[0];
    const float* nf   = (const float*)d_in[1];
    const float* Qt   = (const float*)d_in[2];
    const float* wchi = (const float*)d_in[3];
    const float* hp   = (const float*)d_in[4];
    const float* st   = (const float*)d_in[5];
    const int*   sp   = (const int*)d_in[6];

    float* out   = (float*)d_out;
    float* out_q = out;                 // N = 32768 charges
    float* out_e = out + QM * QN;       // 64 energies

    float* A   = (float*)d_ws;                          // 64*528*528 floats
    float* chi = A + (size_t)QM * NAUG * NAUG;          // 32768 floats
    float* rhs = chi + (size_t)QM * QN;                 // 64*528 floats

    const int N = QM * QN;

    qeq_chi_kernel<<<N / 8, 256, 0, stream>>>(nf, wchi, chi, N);

    dim3 gA((NAUG * NAUG + 255) / 256, QM);
    qeq_build_A_kernel<<<gA, 256, 0, stream>>>(pos, hp, st, sp, A);

    dim3 gR((NAUG + 255) / 256, QM);
    qeq_build_rhs_kernel<<<gR, 256, 0, stream>>>(chi, Qt, rhs);

    qeq_lu_solve_kernel<<<QM, 256, 0, stream>>>(A, rhs, chi, Qt, out_q, out_e);
}